// GAT_66022237274338
// MI455X (gfx1250) — compile-verified
//
#include <hip/hip_runtime.h>
#include <hip/hip_bf16.h>

// ---------------------------------------------------------------------------
// GAT 2-layer forward for gfx1250 (MI455X).
// fp32 throughout (atomic-bound workload; accumulators live in 192MB L2).
// Layer-2 projection [N,128]x[128,4] runs on V_WMMA_F32_16X16X4_F32.
// ---------------------------------------------------------------------------

typedef __attribute__((ext_vector_type(2))) float v2f;
typedef __attribute__((ext_vector_type(8))) float v8f;

#define NEG_SLOPE 0.2f
#define EPS_SM    1e-16f
#define ENC_NEGINF 0x007FFFFFu   // enc(-inf)

// Monotonic order-preserving float<->uint mapping for atomicMax on floats.
__device__ __forceinline__ unsigned encf(float f) {
    unsigned u = __float_as_uint(f);
    return (u & 0x80000000u) ? ~u : (u | 0x80000000u);
}
__device__ __forceinline__ float decf(unsigned u) {
    return (u & 0x80000000u) ? __uint_as_float(u & 0x7FFFFFFFu)
                             : __uint_as_float(~u);
}
__device__ __forceinline__ float leaky(float e) {
    return e > 0.0f ? e : NEG_SLOPE * e;
}

// ---------------------------------------------------------------------------
// Utility fills
// ---------------------------------------------------------------------------
__global__ void fill_u32(unsigned* __restrict__ p, int n, unsigned v) {
    int i = blockIdx.x * blockDim.x + threadIdx.x;
    if (i < n) p[i] = v;
}

// Pre-pack W2 [128,4] into the exact per-lane WMMA B-operand layout:
//   w2b[iter][lane][p], iter = K/4 step (0..31), lane = wave lane, p = 0/1.
//   lane<16  -> covers K = iter*4 + {0,1}, N = lane
//   lane>=16 -> covers K = iter*4 + {2,3}, N = lane-16
// Columns N>=4 are zero padding. Each lane then does ONE contiguous b64 load
// per K-step and the wave's 256B request is fully coalesced.
__global__ void pack_w2b(const float* __restrict__ W2, float* __restrict__ w2b) {
    int i = blockIdx.x * blockDim.x + threadIdx.x;   // 0..2047 = 32*32*2
    if (i >= 32 * 32 * 2) return;
    int p    = i & 1;
    int j    = i >> 1;          // iter*32 + lane
    int lane = j & 31;
    int iter = j >> 5;
    int half = lane >> 4, l16 = lane & 15;
    int k = iter * 4 + 2 * half + p;
    w2b[i] = (l16 < 4) ? W2[k * 4 + l16] : 0.0f;
}

// ---------------------------------------------------------------------------
// Layer 1: h1 = x @ W1 (K=2), per-head attention dots. One block per node,
// 128 threads = 128 channels. W1/att vectors staged in LDS.
// ---------------------------------------------------------------------------
__global__ void node_layer1(const float* __restrict__ x,
                            const float* __restrict__ W1,     // [2,128]
                            const float* __restrict__ attS,   // [8,16]
                            const float* __restrict__ attD,   // [8,16]
                            float* __restrict__ h1,           // [N,128]
                            float* __restrict__ aS1,          // [N,8]
                            float* __restrict__ aD1,          // [N,8]
                            int N) {
    __shared__ float sW0[128], sW1v[128], sAS[128], sAD[128], rS[128], rD[128];
    int tid = threadIdx.x;
    sW0[tid]  = W1[tid];
    sW1v[tid] = W1[128 + tid];
    sAS[tid]  = attS[tid];
    sAD[tid]  = attD[tid];
    __syncthreads();
    int n = blockIdx.x;
    if (n >= N) return;                        // uniform per block
    float x0 = x[n * 2], x1 = x[n * 2 + 1];
    float v = fmaf(x1, sW1v[tid], x0 * sW0[tid]);
    h1[n * 128 + tid] = v;
    rS[tid] = v * sAS[tid];
    rD[tid] = v * sAD[tid];
    __syncthreads();
    if (tid < 8) {
        float a = 0.0f, b = 0.0f;
        #pragma unroll
        for (int cc = 0; cc < 16; ++cc) { a += rS[tid * 16 + cc]; b += rD[tid * 16 + cc]; }
        aS1[n * 8 + tid] = a;
        aD1[n * 8 + tid] = b;
    }
}

// ---------------------------------------------------------------------------
// Segment-softmax pass 1: per-edge leaky score -> atomicMax into m[dst].
// ---------------------------------------------------------------------------
__global__ void edge_max1(const int* __restrict__ ei, int E,
                          const float* __restrict__ aS1, const float* __restrict__ aD1,
                          unsigned* __restrict__ m1) {
    int e = blockIdx.x * blockDim.x + threadIdx.x;
    if (e >= E) return;
    int src = ei[e], dst = ei[E + e];
    #pragma unroll
    for (int h = 0; h < 8; ++h) {
        float ee = leaky(aS1[src * 8 + h] + aD1[dst * 8 + h]);
        atomicMax(&m1[dst * 8 + h], encf(ee));
    }
}

// Segment-softmax pass 2: exp(e - m[dst]) -> atomicAdd into s[dst].
__global__ void edge_sum1(const int* __restrict__ ei, int E,
                          const float* __restrict__ aS1, const float* __restrict__ aD1,
                          const unsigned* __restrict__ m1, float* __restrict__ s1) {
    int e = blockIdx.x * blockDim.x + threadIdx.x;
    if (e >= E) return;
    int src = ei[e], dst = ei[E + e];
    #pragma unroll
    for (int h = 0; h < 8; ++h) {
        float ee = leaky(aS1[src * 8 + h] + aD1[dst * 8 + h]);
        float ex = __expf(ee - decf(m1[dst * 8 + h]));
        atomicAdd(&s1[dst * 8 + h], ex);
    }
}

// Message scatter: 128 lanes per edge (2 edges per 256-thread block).
// Coalesced 512B gather of h1[src], coalesced 512B atomic scatter to acc1[dst].
__global__ void edge_scatter1(const int* __restrict__ ei, int E,
                              const float* __restrict__ aS1, const float* __restrict__ aD1,
                              const unsigned* __restrict__ m1, const float* __restrict__ s1,
                              const float* __restrict__ h1, float* __restrict__ acc1) {
    __shared__ float sAl[2][8];
    __shared__ int   sSrc[2], sDst[2];
    int local = threadIdx.x & 127;
    int eo    = threadIdx.x >> 7;
    int e     = blockIdx.x * 2 + eo;
    if (e < E && local < 8) {
        int src = ei[e], dst = ei[E + e];
        if (local == 0) { sSrc[eo] = src; sDst[eo] = dst; }
        int h = local;
        float ee = leaky(aS1[src * 8 + h] + aD1[dst * 8 + h]);
        float ex = __expf(ee - decf(m1[dst * 8 + h]));
        sAl[eo][h] = ex / (s1[dst * 8 + h] + EPS_SM);
    }
    __syncthreads();
    if (e < E) {
        int src = sSrc[eo], dst = sDst[eo];
        float v = h1[src * 128 + local] * sAl[eo][local >> 4];
        atomicAdd(&acc1[dst * 128 + local], v);
    }
}

// bias + ReLU in place; acc1 becomes the layer-2 input.
__global__ void finalize1(float* __restrict__ acc1, const float* __restrict__ b1, int total) {
    int i = blockIdx.x * blockDim.x + threadIdx.x;
    if (i >= total) return;
    float v = acc1[i] + b1[i & 127];
    acc1[i] = v > 0.0f ? v : 0.0f;
}

// ---------------------------------------------------------------------------
// Layer-2 GEMM via V_WMMA_F32_16X16X4_F32.
// One wave per 16-node tile; K=128 swept in 32 steps of 4; N padded 4->16.
// A (16x4 f32): lane<16 -> M=lane, K={0,1}; lane>=16 -> M=lane-16, K={2,3}.
// B comes pre-packed in per-lane layout (one coalesced b64 per lane per step).
// D (16x16 f32): VGPR r: lanes0-15 -> M=r, lanes16-31 -> M=8+r, N=lane%16.
// Out-of-range rows are CLAMPED (not masked): their results are garbage but
// are never stored, so the K-loop stays divergence-free and EXEC stays full
// around the WMMA as the ISA requires.
// ---------------------------------------------------------------------------
__global__ void gemm2_wmma(const float* __restrict__ rh1,  // [N,128] relu(h1)
                           const float* __restrict__ w2b,  // packed [32][32][2]
                           float* __restrict__ h2,         // [N,4]
                           int nTiles, int N) {
    int wave = blockIdx.x * (blockDim.x >> 5) + (threadIdx.x >> 5);
    if (wave >= nTiles) return;                // wave-uniform exit, EXEC stays full
    int lane = threadIdx.x & 31;
    int half = lane >> 4;                      // 0: K 0/1 group, 1: K 2/3 group
    int l16  = lane & 15;
    int row  = wave * 16 + l16;
    int rowc = row < N ? row : N - 1;          // clamp: no masked loads needed
    const float* arow = rh1 + (size_t)rowc * 128 + 2 * half;
    const float* brow = w2b + lane * 2;

    v8f acc = {};
    #pragma unroll
    for (int it = 0; it < 32; ++it) {
        v2f a = *reinterpret_cast<const v2f*>(arow + it * 4);   // global_load_b64
        v2f b = *reinterpret_cast<const v2f*>(brow + it * 64);  // global_load_b64
        acc = __builtin_amdgcn_wmma_f32_16x16x4_f32(
            /*neg_a=*/false, a, /*neg_b=*/false, b,
            /*c_mod=*/(short)0, acc, /*reuse_a=*/false, /*reuse_b=*/false);
    }

    if (l16 < 4) {                             // only 4 valid output columns
        int base = wave * 16 + half * 8;
        #pragma unroll
        for (int r = 0; r < 8; ++r) {
            int node = base + r;
            if (node < N) h2[node * 4 + l16] = acc[r];
        }
    }
}

// Per-node attention dots for layer 2 (1 head, 4 channels).
__global__ void attn2(const float* __restrict__ h2,
                      const float* __restrict__ attS2, const float* __restrict__ attD2,
                      float* __restrict__ aS2, float* __restrict__ aD2, int N) {
    int n = blockIdx.x * blockDim.x + threadIdx.x;
    if (n >= N) return;
    float s = 0.0f, d = 0.0f;
    #pragma unroll
    for (int c = 0; c < 4; ++c) {
        float v = h2[n * 4 + c];
        s += v * attS2[c];
        d += v * attD2[c];
    }
    aS2[n] = s; aD2[n] = d;
}

__global__ void edge_max2(const int* __restrict__ ei, int E,
                          const float* __restrict__ aS2, const float* __restrict__ aD2,
                          unsigned* __restrict__ m2) {
    int e = blockIdx.x * blockDim.x + threadIdx.x;
    if (e >= E) return;
    float ee = leaky(aS2[ei[e]] + aD2[ei[E + e]]);
    atomicMax(&m2[ei[E + e]], encf(ee));
}

__global__ void edge_sum2(const int* __restrict__ ei, int E,
                          const float* __restrict__ aS2, const float* __restrict__ aD2,
                          const unsigned* __restrict__ m2, float* __restrict__ s2) {
    int e = blockIdx.x * blockDim.x + threadIdx.x;
    if (e >= E) return;
    int src = ei[e], dst = ei[E + e];
    float ee = leaky(aS2[src] + aD2[dst]);
    atomicAdd(&s2[dst], __expf(ee - decf(m2[dst])));
}

__global__ void edge_scatter2(const int* __restrict__ ei, int E,
                              const float* __restrict__ aS2, const float* __restrict__ aD2,
                              const unsigned* __restrict__ m2, const float* __restrict__ s2,
                              const float* __restrict__ h2, float* __restrict__ acc2) {
    int e = blockIdx.x * blockDim.x + threadIdx.x;
    if (e >= E) return;
    int src = ei[e], dst = ei[E + e];
    float ee = leaky(aS2[src] + aD2[dst]);
    float ex = __expf(ee - decf(m2[dst]));
    float al = ex / (s2[dst] + EPS_SM);
    #pragma unroll
    for (int c = 0; c < 4; ++c)
        atomicAdd(&acc2[dst * 4 + c], h2[src * 4 + c] * al);
}

// bias + log_softmax over the 4 classes -> d_out.
__global__ void final_logsoftmax(const float* __restrict__ acc2,
                                 const float* __restrict__ b2,
                                 float* __restrict__ out, int N) {
    int n = blockIdx.x * blockDim.x + threadIdx.x;
    if (n >= N) return;
    float z[4], mx = -3.0e38f;
    #pragma unroll
    for (int c = 0; c < 4; ++c) { z[c] = acc2[n * 4 + c] + b2[c]; mx = fmaxf(mx, z[c]); }
    float s = 0.0f;
    #pragma unroll
    for (int c = 0; c < 4; ++c) s += __expf(z[c] - mx);
    float ls = __logf(s);
    #pragma unroll
    for (int c = 0; c < 4; ++c) out[n * 4 + c] = z[c] - mx - ls;
}

// ---------------------------------------------------------------------------
// Host launcher
// ---------------------------------------------------------------------------
static inline int cdiv(int a, int b) { return (a + b - 1) / b; }

extern "C" void kernel_launch(void* const* d_in, const int* in_sizes, int n_in,
                              void* d_out, int out_size, void* d_ws, size_t ws_size,
                              hipStream_t stream) {
    const float* x     = (const float*)d_in[0];   // [N,2]
    const int*   ei    = (const int*)  d_in[1];   // [2,E]
    const float* W1    = (const float*)d_in[2];   // [2,128]
    const float* attS1 = (const float*)d_in[3];   // [8,16]
    const float* attD1 = (const float*)d_in[4];   // [8,16]
    const float* b1    = (const float*)d_in[5];   // [128]
    const float* W2    = (const float*)d_in[6];   // [128,4]
    const float* attS2 = (const float*)d_in[7];   // [1,4]
    const float* attD2 = (const float*)d_in[8];   // [1,4]
    const float* b2    = (const float*)d_in[9];   // [4]
    float* out = (float*)d_out;

    const int N = in_sizes[0] / 2;
    const int E = in_sizes[1] / 2;

    // Workspace carve-up (fp32 elements)
    float* ws   = (float*)d_ws;
    float*    h1   = ws;                       // N*128
    float*    aS1  = h1   + (size_t)N * 128;   // N*8
    float*    aD1  = aS1  + (size_t)N * 8;     // N*8
    unsigned* m1   = (unsigned*)(aD1 + (size_t)N * 8);   // N*8
    float*    s1   = (float*)(m1 + (size_t)N * 8);       // N*8
    float*    acc1 = s1   + (size_t)N * 8;     // N*128 (becomes relu input to L2)
    float*    h2   = acc1 + (size_t)N * 128;   // N*4
    float*    aS2  = h2   + (size_t)N * 4;     // N
    float*    aD2  = aS2  + (size_t)N;         // N
    unsigned* m2   = (unsigned*)(aD2 + (size_t)N);       // N
    float*    s2   = (float*)(m2 + (size_t)N);           // N
    float*    acc2 = s2   + (size_t)N;         // N*4
    float*    w2b  = acc2 + (size_t)N * 4;     // 32*32*2 = 2048 (packed B)

    const int B = 256;

    // --- init ---
    fill_u32<<<cdiv(N * 8, B),   B, 0, stream>>>(m1, N * 8, ENC_NEGINF);
    fill_u32<<<cdiv(N * 8, B),   B, 0, stream>>>((unsigned*)s1, N * 8, 0u);
    fill_u32<<<cdiv(N * 128, B), B, 0, stream>>>((unsigned*)acc1, N * 128, 0u);
    fill_u32<<<cdiv(N, B),       B, 0, stream>>>(m2, N, ENC_NEGINF);
    fill_u32<<<cdiv(N, B),       B, 0, stream>>>((unsigned*)s2, N, 0u);
    fill_u32<<<cdiv(N * 4, B),   B, 0, stream>>>((unsigned*)acc2, N * 4, 0u);
    pack_w2b<<<cdiv(2048, B),    B, 0, stream>>>(W2, w2b);

    // --- layer 1 ---
    node_layer1<<<N, 128, 0, stream>>>(x, W1, attS1, attD1, h1, aS1, aD1, N);
    edge_max1<<<cdiv(E, B), B, 0, stream>>>(ei, E, aS1, aD1, m1);
    edge_sum1<<<cdiv(E, B), B, 0, stream>>>(ei, E, aS1, aD1, m1, s1);
    edge_scatter1<<<cdiv(E, 2), 256, 0, stream>>>(ei, E, aS1, aD1, m1, s1, h1, acc1);
    finalize1<<<cdiv(N * 128, B), B, 0, stream>>>(acc1, b1, N * 128);

    // --- layer 2 (WMMA GEMM) ---
    const int nTiles = cdiv(N, 16);
    gemm2_wmma<<<cdiv(nTiles, 8), 256, 0, stream>>>(acc1, w2b, h2, nTiles, N);
    attn2<<<cdiv(N, B), B, 0, stream>>>(h2, attS2, attD2, aS2, aD2, N);
    edge_max2<<<cdiv(E, B), B, 0, stream>>>(ei, E, aS2, aD2, m2);
    edge_sum2<<<cdiv(E, B), B, 0, stream>>>(ei, E, aS2, aD2, m2, s2);
    edge_scatter2<<<cdiv(E, B), B, 0, stream>>>(ei, E, aS2, aD2, m2, s2, h2, acc2);
    final_logsoftmax<<<cdiv(N, B), B, 0, stream>>>(acc2, b2, out, N);

    (void)n_in; (void)out_size; (void)ws_size;
}